// LiteLinear_64845416235340
// MI455X (gfx1250) — compile-verified
//
#include <hip/hip_runtime.h>
#include <stdint.h>

// ---------------------------------------------------------------------------
// LoRA linear: out = x @ W^T + bias + scaling[a] * (x @ A[a]^T) @ B[a]^T
//
// Base GEMM (137 GFLOP fp32, compute-bound on MI455X) runs on the fp32 matrix
// pipe via V_WMMA_F32_16X16X4_F32. Data movement: both the x tile and the W
// tile are plain row-major 2D copies (the B fragment reads consecutive K for
// fixed N, so W is staged UNtransposed), which lets the Tensor Data Mover
// (tensor_load_to_lds, TENSORcnt) do all global->LDS staging, double-buffered
// against the WMMA loop. TDM pad_enable (16 dwords + 4 pad) gives the 20-float
// row pitch that keeps ds_load_b64 fragment reads bank-conflict-free.
// ---------------------------------------------------------------------------

typedef float v2f  __attribute__((ext_vector_type(2)));
typedef float v8f  __attribute__((ext_vector_type(8)));
typedef unsigned int u32x4 __attribute__((ext_vector_type(4)));
typedef int   i32x4 __attribute__((ext_vector_type(4)));
typedef int   i32x8 __attribute__((ext_vector_type(8)));

#define DIN  2048
#define DOUT 2048
#define RANK 16

// Block tile 128(M) x 256(N), 8 waves as 2x4, each wave 64x64 = 4x4 WMMA tiles.
constexpr int BM  = 128;
constexpr int BN  = 256;
constexpr int BK  = 16;
constexpr int BKP = 20;   // row pitch in floats: 16 data + 4 pad (80B, 16B-aligned)

#if defined(__gfx1250__) && __has_builtin(__builtin_amdgcn_tensor_load_to_lds) && \
    __has_builtin(__builtin_amdgcn_s_wait_tensorcnt)
#define USE_TDM 1
#else
#define USE_TDM 0
#endif

#if USE_TDM
// Issue one TDM descriptor: copy a (rows x 16) f32 tile, row stride DIN
// elements, from global to LDS with 4-dword padding every 16 dwords.
// D# packing per CDNA5 ISA 8.3 (group0) / 8.4 (group1); groups 2-3 and the
// trailing group are unused for a 2D tile (zero-filled).
__device__ __forceinline__ void tdm_load_tile(uint32_t lds_addr,
                                              const float* gptr,
                                              uint32_t rows,
                                              uint32_t tensor_rows) {
    const uint64_t ga = (uint64_t)(uintptr_t)gptr;
    u32x4 g0;
    g0[0] = 1u;                                   // count=1 (valid user D#)
    g0[1] = lds_addr;                             // lds_addr [63:32]
    g0[2] = (uint32_t)ga;                         // global_addr lo
    g0[3] = (uint32_t)((ga >> 32) & 0x1FFFFFFu)   // global_addr [56:32]
          | (2u << 30);                           // type=2 ("image")
    const uint32_t td0 = DIN;                     // tensor_dim0 (elements)
    const uint32_t td1 = tensor_rows;             // tensor_dim1
    const uint32_t t0  = BK;                      // tile_dim0 = 16 elements
    i32x8 g1;
    // data_size=4B (2<<16) | pad_enable (1<<20) | pad_interval=16dw (3<<22)
    // | pad_amount=4dw (3<<25); workgroup_mask=0 (not in a cluster)
    g1[0] = (int)((2u << 16) | (1u << 20) | (3u << 22) | (3u << 25));
    g1[1] = (int)((td0 & 0xFFFFu) << 16);                  // dim0 lo16 @ [63:48]
    g1[2] = (int)((td0 >> 16) | ((td1 & 0xFFFFu) << 16));  // dim0 hi | dim1 lo
    g1[3] = (int)((td1 >> 16) | (t0 << 16));               // dim1 hi | tile_dim0
    g1[4] = (int)rows;                                     // tile_dim1 (tile_dim2=0)
    g1[5] = (int)DIN;                                      // tensor_dim0_stride lo32
    g1[6] = 0;                                             // stride0 hi | stride1 lo
    g1[7] = 0;
    const i32x4 z4 = {0, 0, 0, 0};
    const i32x8 z8 = {0, 0, 0, 0, 0, 0, 0, 0};
    __builtin_amdgcn_tensor_load_to_lds(g0, g1, z4, z4, z8, /*cpol=*/0);
}
#endif

__global__ __launch_bounds__(256)
void lora_base_gemm_wmma(const float* __restrict__ X,
                         const float* __restrict__ W,
                         const float* __restrict__ bias,
                         float* __restrict__ out, int M)
{
    __shared__ float As[2][BM * BKP];   // x tile,  As[m][k] row-major (padded)
    __shared__ float Bs[2][BN * BKP];   // W tile,  Bs[n][k] row-major (padded)

    const int tid   = threadIdx.x;
    const int lane  = tid & 31;
    const int wave  = tid >> 5;        // 0..7
    const int wm    = wave >> 2;       // 0..1
    const int wn    = wave & 3;        // 0..3
    const int lhalf = lane >> 4;
    const int l16   = lane & 15;

    const int block_m = blockIdx.y * BM;
    const int block_n = blockIdx.x * BN;
    const int wrow = wm * 64;
    const int wcol = wn * 64;

    // Accumulators seeded with bias (WMMA computes A*B + C).
    v8f acc[4][4];
    #pragma unroll
    for (int j = 0; j < 4; ++j) {
        const float bv = bias[block_n + wcol + j * 16 + l16];
        #pragma unroll
        for (int i = 0; i < 4; ++i)
            #pragma unroll
            for (int e = 0; e < 8; ++e)
                acc[i][j][e] = bv;
    }

    const int nchunks = DIN / BK;      // 128

#if USE_TDM
    const uint32_t ldsA[2] = { (uint32_t)(uintptr_t)&As[0][0],
                               (uint32_t)(uintptr_t)&As[1][0] };
    const uint32_t ldsB[2] = { (uint32_t)(uintptr_t)&Bs[0][0],
                               (uint32_t)(uintptr_t)&Bs[1][0] };
    const float* Xt = X + (size_t)block_m * DIN;
    const float* Wt = W + (size_t)block_n * DIN;

    if (wave == 0) {                   // TDM ignores EXEC; issue once per WG
        tdm_load_tile(ldsA[0], Xt, BM, (uint32_t)M);
        tdm_load_tile(ldsB[0], Wt, BN, DOUT);
    }

    for (int c = 0; c < nchunks; ++c) {
        const int buf = c & 1;
        if (wave == 0) {
            if (c + 1 < nchunks) {     // prefetch next chunk into other buffer
                const int k0n = (c + 1) * BK;
                tdm_load_tile(ldsA[buf ^ 1], Xt + k0n, BM, (uint32_t)M);
                tdm_load_tile(ldsB[buf ^ 1], Wt + k0n, BN, DOUT);
                __builtin_amdgcn_s_wait_tensorcnt(2);  // current chunk landed
            } else {
                __builtin_amdgcn_s_wait_tensorcnt(0);
            }
        }
        __syncthreads();

        const float* Ab = &As[buf][0];
        const float* Bb = &Bs[buf][0];
        #pragma unroll
        for (int ks = 0; ks < BK; ks += 4) {
            v2f af[4], bf[4];
            #pragma unroll
            for (int i = 0; i < 4; ++i)
                af[i] = *(const v2f*)(Ab + (wrow + i * 16 + l16) * BKP + ks + lhalf * 2);
            #pragma unroll
            for (int j = 0; j < 4; ++j)
                bf[j] = *(const v2f*)(Bb + (wcol + j * 16 + l16) * BKP + ks + lhalf * 2);
            #pragma unroll
            for (int i = 0; i < 4; ++i)
                #pragma unroll
                for (int j = 0; j < 4; ++j)
                    acc[i][j] = __builtin_amdgcn_wmma_f32_16x16x4_f32(
                        false, af[i], false, bf[j], (short)0, acc[i][j], false, false);
        }
        __syncthreads();
    }
#else
    // Fallback: cooperative per-thread staging (single buffer).
    for (int c = 0; c < nchunks; ++c) {
        const int k0 = c * BK;
        {   // A: 128 rows x 16 f32; thread -> row tid/2, 8 floats at (tid&1)*8
            const int r = tid >> 1, cc = (tid & 1) * 8;
            const float* p = X + (size_t)(block_m + r) * DIN + k0 + cc;
            *(float4*)(&As[0][r * BKP + cc])     = *(const float4*)(p);
            *(float4*)(&As[0][r * BKP + cc + 4]) = *(const float4*)(p + 4);
        }
        {   // B: 256 rows x 16 f32; thread -> row tid, 16 floats
            const float* p = W + (size_t)(block_n + tid) * DIN + k0;
            #pragma unroll
            for (int q = 0; q < 4; ++q)
                *(float4*)(&Bs[0][tid * BKP + q * 4]) = *(const float4*)(p + q * 4);
        }
        __syncthreads();
        #pragma unroll
        for (int ks = 0; ks < BK; ks += 4) {
            v2f af[4], bf[4];
            #pragma unroll
            for (int i = 0; i < 4; ++i)
                af[i] = *(const v2f*)(&As[0][(wrow + i * 16 + l16) * BKP + ks + lhalf * 2]);
            #pragma unroll
            for (int j = 0; j < 4; ++j)
                bf[j] = *(const v2f*)(&Bs[0][(wcol + j * 16 + l16) * BKP + ks + lhalf * 2]);
            #pragma unroll
            for (int i = 0; i < 4; ++i)
                #pragma unroll
                for (int j = 0; j < 4; ++j)
                    acc[i][j] = __builtin_amdgcn_wmma_f32_16x16x4_f32(
                        false, af[i], false, bf[j], (short)0, acc[i][j], false, false);
        }
        __syncthreads();
    }
#endif

    // Epilogue: C layout -> row = e + 8*lhalf (+16*i), col = l16 (+16*j).
    #pragma unroll
    for (int i = 0; i < 4; ++i) {
        #pragma unroll
        for (int j = 0; j < 4; ++j) {
            const int col = block_n + wcol + j * 16 + l16;
            #pragma unroll
            for (int e = 0; e < 8; ++e) {
                const int row = block_m + wrow + i * 16 + e + lhalf * 8;
                out[(size_t)row * DOUT + col] = acc[i][j][e];
            }
        }
    }
}

// ---------------------------------------------------------------------------
// LoRA delta: one 256-thread block per token; aid==0 -> skip.
// ---------------------------------------------------------------------------
__global__ __launch_bounds__(256)
void lora_delta_kernel(const float* __restrict__ X,
                       const int*   __restrict__ mapping,
                       const float* __restrict__ la,
                       const float* __restrict__ lb,
                       const float* __restrict__ scaling,
                       float* __restrict__ out)
{
    const int token = blockIdx.x;
    const int aid = mapping[token];
    if (aid == 0) return;
    const int a = aid - 1;

    __shared__ float xs[DIN];
    __shared__ float us[RANK];

    const int tid = threadIdx.x;
    for (int i = tid; i < DIN / 4; i += 256)
        ((float4*)xs)[i] = ((const float4*)(X + (size_t)token * DIN))[i];
    if (tid < RANK) us[tid] = 0.0f;
    __syncthreads();

    const int r     = tid & 15;
    const int chunk = tid >> 4;
    const float* ap = la + ((size_t)a * RANK + r) * DIN + chunk * 128;
    const float* xp = xs + chunk * 128;
    float p = 0.0f;
    #pragma unroll 8
    for (int d = 0; d < 128; d += 4) {
        const float4 xv = *(const float4*)(xp + d);
        const float4 av = *(const float4*)(ap + d);
        p += xv.x * av.x + xv.y * av.y + xv.z * av.z + xv.w * av.w;
    }
    atomicAdd(&us[r], p);
    __syncthreads();

    const float s = scaling[a];
    for (int o = tid; o < DOUT; o += 256) {
        const float* bp = lb + ((size_t)a * DOUT + o) * RANK;
        float acc = 0.0f;
        #pragma unroll
        for (int rr = 0; rr < RANK; ++rr) acc += us[rr] * bp[rr];
        out[(size_t)token * DOUT + o] += s * acc;
    }
}

// ---------------------------------------------------------------------------
extern "C" void kernel_launch(void* const* d_in, const int* in_sizes, int n_in,
                              void* d_out, int out_size, void* d_ws, size_t ws_size,
                              hipStream_t stream) {
    const float* x       = (const float*)d_in[0];
    const int*   mapping = (const int*)  d_in[1];
    const float* weight  = (const float*)d_in[2];
    const float* bias    = (const float*)d_in[3];
    const float* lora_a  = (const float*)d_in[4];
    const float* lora_b  = (const float*)d_in[5];
    const float* scaling = (const float*)d_in[6];
    float* out = (float*)d_out;

    const int M = in_sizes[0] / DIN;          // 16384 tokens

    dim3 grid(DOUT / BN, M / BM);             // (8, 128)
    lora_base_gemm_wmma<<<grid, 256, 0, stream>>>(x, weight, bias, out, M);
    lora_delta_kernel<<<M, 256, 0, stream>>>(x, mapping, lora_a, lora_b, scaling, out);
}